// FixedGaussianBlur_627065225249
// MI455X (gfx1250) — compile-verified
//
#include <hip/hip_runtime.h>
#include <math.h>

// Fixed 21x21 Gaussian blur (sigma=3), separable, reflect padding.
// Input : x  [32, 3, 512, 512] fp32 ; Output: y same shape fp32.
//
// Strategy: conv == matmul against a banded Gaussian matrix.
//   Pass1 (horizontal): D = A(data 16x36) * B(band 36x16), 9x K=4 f32 WMMA
//   Pass2 (vertical)  : D = A(band 16x36) * B(data 36x16), 9x K=4 f32 WMMA
// The banded weight pairs depend only on (lane, chunk) -> precomputed once
// by an init kernel into a 9x32 float2 table, loaded into registers per wave.
// Interior tiles (30/32 per row) need no reflect: pure load+wmma inner loop.

#define KS   21
#define PAD  10
#define HH   512
#define WW   512
#define NPLANES 96            // 32 batches * 3 channels
#define TILES_PER_PLANE 1024  // (512/16)^2
#define WAVES_PER_BLOCK 8
#define NCHUNK 9              // K = 36 = 9 * 4
#define TMP_ELEMS ((size_t)NPLANES * HH * WW)

typedef __attribute__((ext_vector_type(2))) float v2f;
typedef __attribute__((ext_vector_type(8))) float v8f;

__device__ __forceinline__ int refl(int i, int n) {
  if (i < 0)  i = -i;
  if (i >= n) i = 2 * n - 2 - i;
  return i;
}

__device__ __forceinline__ v8f wmma_f32(v2f a, v2f b, v8f c) {
  return __builtin_amdgcn_wmma_f32_16x16x4_f32(
      false, a, false, b, (short)0, c, false, false);
}

// ---------------- Init: build the banded weight-pair table ----------------
// wtab[j*32 + lane] = { g[4j + kb - ln], g[4j + kb + 1 - ln] } (0 outside band)
// where ln = lane&15, kb = (lane>>4)*2. Identical table serves pass1-B and
// pass2-A because the f32 A(16x4) and B(4x16) layouts stripe (M|N, K) the
// same way across lanes.
__global__ __launch_bounds__(288) void gauss_init(v2f* __restrict__ wtab) {
  __shared__ double dge[KS];
  __shared__ float  gs[KS];
  const int t = threadIdx.x;
  if (t < KS) {
    double xx = (double)t - 10.0;
    dge[t] = exp(-(xx * xx) / 18.0);  // 2*sigma^2 = 18
  }
  __syncthreads();
  if (t < KS) {
    double s = 0.0;
#pragma unroll
    for (int i = 0; i < KS; ++i) s += dge[i];
    gs[t] = (float)(dge[t] / s);
  }
  __syncthreads();
  if (t < NCHUNK * 32) {
    const int lane = t & 31;
    const int j    = t >> 5;
    const int ln   = lane & 15;
    const int kb   = (lane >> 4) << 1;
    const int i0   = 4 * j + kb - ln;
    v2f w;
    w.x = (((unsigned)i0)       < (unsigned)KS) ? gs[((unsigned)i0       < (unsigned)KS) ? i0     : 0] : 0.0f;
    w.y = (((unsigned)(i0 + 1)) < (unsigned)KS) ? gs[((unsigned)(i0 + 1) < (unsigned)KS) ? i0 + 1 : 0] : 0.0f;
    wtab[t] = w;
  }
}

// ---------------- Pass 1: horizontal blur  x -> tmp ----------------
__global__ __launch_bounds__(256) void gauss_h(const float* __restrict__ x,
                                               float* __restrict__ tmp,
                                               const v2f* __restrict__ wtab) {
  const int wave = threadIdx.x >> 5;
  const int lane = threadIdx.x & 31;
  const int tile = blockIdx.x * WAVES_PER_BLOCK + wave;
  const int p  = tile >> 10;
  const int t  = tile & (TILES_PER_PLANE - 1);
  const int h0 = (t >> 5) << 4;
  const int w0 = (t & 31) << 4;

  const int ln = lane & 15;
  const int kb = (lane >> 4) << 1;

  // weight pairs -> registers (L2-hot broadcast loads)
  v2f w[NCHUNK];
#pragma unroll
  for (int j = 0; j < NCHUNK; ++j) w[j] = wtab[j * 32 + lane];

  const float* __restrict__ rowp =
      x + (size_t)p * (HH * WW) + (size_t)(h0 + ln) * WW;

  v8f acc = {};
  if (w0 >= PAD && w0 + 16 + PAD <= WW) {
    // interior: contiguous aligned pair loads at static offsets
    const float* __restrict__ p0 = rowp + (w0 - PAD + kb);
#pragma unroll
    for (int j = 0; j < NCHUNK; ++j) {
      v2f a = *(const v2f*)(p0 + 4 * j);
      acc = wmma_f32(a, w[j], acc);
    }
  } else {
    // border: per-element reflect
#pragma unroll
    for (int j = 0; j < NCHUNK; ++j) {
      const int cc = w0 - PAD + 4 * j + kb;
      v2f a;
      a.x = rowp[refl(cc,     WW)];
      a.y = rowp[refl(cc + 1, WW)];
      acc = wmma_f32(a, w[j], acc);
    }
  }

  // D: col = lane&15, rows = (lane>>4)*8 + v
  const int rb = h0 + ((lane >> 4) << 3);
  float* __restrict__ dst =
      tmp + (size_t)p * (HH * WW) + (size_t)rb * WW + w0 + ln;
#pragma unroll
  for (int v = 0; v < 8; ++v) dst[(size_t)v * WW] = acc[v];
}

// ---------------- Pass 2: vertical blur  tmp -> out ----------------
__global__ __launch_bounds__(256) void gauss_v(const float* __restrict__ tmp,
                                               float* __restrict__ out,
                                               const v2f* __restrict__ wtab) {
  const int wave = threadIdx.x >> 5;
  const int lane = threadIdx.x & 31;
  const int tile = blockIdx.x * WAVES_PER_BLOCK + wave;
  const int p  = tile >> 10;
  const int t  = tile & (TILES_PER_PLANE - 1);
  const int h0 = (t >> 5) << 4;
  const int w0 = (t & 31) << 4;

  const int ln = lane & 15;
  const int kb = (lane >> 4) << 1;

  v2f w[NCHUNK];
#pragma unroll
  for (int j = 0; j < NCHUNK; ++j) w[j] = wtab[j * 32 + lane];

  const float* __restrict__ src = tmp + (size_t)p * (HH * WW);

  v8f acc = {};
  if (h0 >= PAD && h0 + 16 + PAD <= HH) {
    // interior: static row-stride offsets from one per-lane base
    const float* __restrict__ p0 =
        src + (size_t)(h0 - PAD + kb) * WW + w0 + ln;
#pragma unroll
    for (int j = 0; j < NCHUNK; ++j) {
      v2f b;
      b.x = p0[(size_t)(4 * j)     * WW];
      b.y = p0[(size_t)(4 * j + 1) * WW];
      acc = wmma_f32(w[j], b, acc);
    }
  } else {
#pragma unroll
    for (int j = 0; j < NCHUNK; ++j) {
      const int rr = h0 - PAD + 4 * j + kb;
      v2f b;
      b.x = src[(size_t)refl(rr,     HH) * WW + w0 + ln];
      b.y = src[(size_t)refl(rr + 1, HH) * WW + w0 + ln];
      acc = wmma_f32(w[j], b, acc);
    }
  }

  const int rb = h0 + ((lane >> 4) << 3);
  float* __restrict__ dst =
      out + (size_t)p * (HH * WW) + (size_t)rb * WW + w0 + ln;
#pragma unroll
  for (int v = 0; v < 8; ++v) dst[(size_t)v * WW] = acc[v];
}

extern "C" void kernel_launch(void* const* d_in, const int* in_sizes, int n_in,
                              void* d_out, int out_size, void* d_ws, size_t ws_size,
                              hipStream_t stream) {
  const float* x = (const float*)d_in[0];
  float* y   = (float*)d_out;
  float* tmp = (float*)d_ws;                 // 32*3*512*512*4 = ~100.7 MB
  v2f*   wtab = (v2f*)(tmp + TMP_ELEMS);     // + 9*32*8 = 2304 B

  const int tiles  = NPLANES * TILES_PER_PLANE;   // 98304
  const int blocks = tiles / WAVES_PER_BLOCK;     // 12288, exact

  gauss_init<<<1, 288, 0, stream>>>(wtab);
  gauss_h<<<blocks, 256, 0, stream>>>(x, tmp, wtab);
  gauss_v<<<blocks, 256, 0, stream>>>(tmp, y, wtab);
}